// SelfAttention3D_2405181686236
// MI455X (gfx1250) — compile-verified
//
#include <hip/hip_runtime.h>
#include <math.h>

#define B_  2
#define C_  64
#define QK_ 8
#define N_  8192   // D*H*W = 8*32*32

typedef float v2f __attribute__((ext_vector_type(2)));
typedef float v8f __attribute__((ext_vector_type(8)));

// ---------------------------------------------------------------------------
// Kernel 1: fused 1x1x1 conv projections q = Wq x + bq, k = Wk x + bk,
//           v = Wv x + bv.  One thread per (b, n) spatial position: load the
//           64-channel column of x once, produce all 80 outputs.
// ---------------------------------------------------------------------------
__global__ __launch_bounds__(256)
void qkv_kernel(const float* __restrict__ x,
                const float* __restrict__ Wq, const float* __restrict__ bq,
                const float* __restrict__ Wk, const float* __restrict__ bk,
                const float* __restrict__ Wv, const float* __restrict__ bv,
                float* __restrict__ q, float* __restrict__ k, float* __restrict__ v)
{
    int t = blockIdx.x * blockDim.x + threadIdx.x;     // 0 .. B*N-1
    if (t >= B_ * N_) return;
    int b = t / N_;
    int n = t % N_;

    const float* xc = x + (size_t)b * C_ * N_ + n;
    float xr[C_];
#pragma unroll
    for (int c = 0; c < C_; ++c) xr[c] = xc[(size_t)c * N_];

#pragma unroll
    for (int o = 0; o < QK_; ++o) {
        float sq = bq[o];
        float sk = bk[o];
#pragma unroll
        for (int c = 0; c < C_; ++c) {
            sq = fmaf(Wq[o * C_ + c], xr[c], sq);
            sk = fmaf(Wk[o * C_ + c], xr[c], sk);
        }
        q[((size_t)b * QK_ + o) * N_ + n] = sq;
        k[((size_t)b * QK_ + o) * N_ + n] = sk;
    }
    for (int o = 0; o < C_; ++o) {
        float sv = bv[o];
#pragma unroll
        for (int c = 0; c < C_; ++c) sv = fmaf(Wv[o * C_ + c], xr[c], sv);
        v[((size_t)b * C_ + o) * N_ + n] = sv;
    }
}

// ---------------------------------------------------------------------------
// Kernel 2: flash attention with V_WMMA_F32_16X16X4_F32.
// One wave32 per 16-row query tile. 8 waves / block.
//
// Energy is computed transposed (E^T = K^T * Q) so that in the WMMA C/D
// layout every lane's 8 result registers belong to ONE query row i = lane%16,
// making the online-softmax row reduction 7 in-register max/adds plus a
// single shfl_xor(16).  O' = V * P^T keeps i on the same lane, so the
// softmax rescale factors are lane-uniform scalars.
// ---------------------------------------------------------------------------
__global__ __launch_bounds__(256)
void attn_kernel(const float* __restrict__ q, const float* __restrict__ k,
                 const float* __restrict__ v, const float* __restrict__ x,
                 const float* __restrict__ gamma, float* __restrict__ out)
{
    __shared__ float ldsP[8][16 * 17];   // per-wave P staging tile (stride 17)

    const int wave = threadIdx.x >> 5;
    const int lane = threadIdx.x & 31;
    const int half = lane >> 4;          // 0: lanes 0-15, 1: lanes 16-31
    const int m    = lane & 15;

    const int tile = blockIdx.x * 8 + wave;        // 0 .. B*(N/16)-1
    if (tile >= B_ * (N_ / 16)) return;
    const int b  = tile / (N_ / 16);
    const int i0 = (tile % (N_ / 16)) * 16;

    const float* Qb = q + (size_t)b * QK_ * N_;
    const float* Kb = k + (size_t)b * QK_ * N_;
    const float* Vb = v + (size_t)b * C_ * N_;

    // B-operand (Q tile, KxN = q x i) — fixed across the j loop.
    // VGPR vg of half h holds contraction index qdim = base + 2*h + vg.
    v2f bq0, bq1;
    bq0.x = Qb[(size_t)(0 + 2 * half) * N_ + i0 + m];
    bq0.y = Qb[(size_t)(1 + 2 * half) * N_ + i0 + m];
    bq1.x = Qb[(size_t)(4 + 2 * half) * N_ + i0 + m];
    bq1.y = Qb[(size_t)(5 + 2 * half) * N_ + i0 + m];

    float mrow = -INFINITY;   // running row max   (row i = i0 + m)
    float lrow = 0.0f;        // running row sum
    v8f o0 = {}, o1 = {}, o2 = {}, o3 = {};   // O'[c,i] tiles, c-tiles 0..3

    float* Pw = ldsP[wave];

    for (int j0 = 0; j0 < N_; j0 += 16) {
        // ---- E^T tile: A-operand = K tile (M=j, K=qdim) -------------------
        v2f ak0, ak1;
        ak0.x = Kb[(size_t)(0 + 2 * half) * N_ + j0 + m];
        ak0.y = Kb[(size_t)(1 + 2 * half) * N_ + j0 + m];
        ak1.x = Kb[(size_t)(4 + 2 * half) * N_ + j0 + m];
        ak1.y = Kb[(size_t)(5 + 2 * half) * N_ + j0 + m];

        v8f e = {};
        e = __builtin_amdgcn_wmma_f32_16x16x4_f32(false, ak0, false, bq0,
                                                  (short)0, e, false, false);
        e = __builtin_amdgcn_wmma_f32_16x16x4_f32(false, ak1, false, bq1,
                                                  (short)0, e, false, false);
        // e[r] = E[i = i0+m, j = j0 + r + 8*half]

        // ---- online softmax update (row i lives in this lane) -------------
        float tmax = e[0];
#pragma unroll
        for (int r = 1; r < 8; ++r) tmax = fmaxf(tmax, e[r]);
        tmax = fmaxf(tmax, __shfl_xor(tmax, 16, 32));
        const float mnew = fmaxf(mrow, tmax);
        const float corr = __expf(mrow - mnew);   // 0 on first iteration

        float p[8];
        float s = 0.0f;
#pragma unroll
        for (int r = 0; r < 8; ++r) {
            p[r] = __expf(e[r] - mnew);
            s += p[r];
        }
        s += __shfl_xor(s, 16, 32);
        lrow = lrow * corr + s;
        mrow = mnew;

        o0 *= corr; o1 *= corr; o2 *= corr; o3 *= corr;

        // ---- stage P into LDS in [j_local][i] layout ----------------------
#pragma unroll
        for (int r = 0; r < 8; ++r)
            Pw[(r + 8 * half) * 17 + m] = p[r];
        // LDS is in-order per wave; fence compiler + drain DS counter.
        asm volatile("s_wait_dscnt 0x0" ::: "memory");

        // ---- O' += V(c x j) * P^T(j x i): 4 K-chunks x 4 c-tiles ---------
#pragma unroll
        for (int jb = 0; jb < 4; ++jb) {
            v2f bp;   // B-operand: P[j_local = jb*4 + 2*half + vg][i = m]
            bp.x = Pw[(jb * 4 + 2 * half + 0) * 17 + m];
            bp.y = Pw[(jb * 4 + 2 * half + 1) * 17 + m];

            const int jg = j0 + jb * 4 + 2 * half;   // even -> 8B aligned
#pragma unroll
            for (int ct = 0; ct < 4; ++ct) {
                const float* vp = Vb + (size_t)(ct * 16 + m) * N_ + jg;
                v2f av;  // A-operand: V[c = ct*16 + m][j = jg + vg]
                av.x = vp[0];
                av.y = vp[1];
                switch (ct) {
                case 0: o0 = __builtin_amdgcn_wmma_f32_16x16x4_f32(false, av, false, bp, (short)0, o0, false, false); break;
                case 1: o1 = __builtin_amdgcn_wmma_f32_16x16x4_f32(false, av, false, bp, (short)0, o1, false, false); break;
                case 2: o2 = __builtin_amdgcn_wmma_f32_16x16x4_f32(false, av, false, bp, (short)0, o2, false, false); break;
                case 3: o3 = __builtin_amdgcn_wmma_f32_16x16x4_f32(false, av, false, bp, (short)0, o3, false, false); break;
                }
            }
        }
        asm volatile("s_wait_dscnt 0x0" ::: "memory"); // reads done before next-iter stores
    }

    // ---- epilogue: out = gamma * (O'/l) + x -------------------------------
    const float linv = 1.0f / lrow;            // lane-uniform per row i
    const float g    = gamma[0];
    const float* xb  = x   + (size_t)b * C_ * N_;
    float*       ob  = out + (size_t)b * C_ * N_;

#pragma unroll
    for (int ct = 0; ct < 4; ++ct) {
        v8f oo = (ct == 0) ? o0 : (ct == 1) ? o1 : (ct == 2) ? o2 : o3;
#pragma unroll
        for (int r = 0; r < 8; ++r) {
            const int c = ct * 16 + r + 8 * half;   // C/D layout row
            const size_t idx = (size_t)c * N_ + i0 + m;
            ob[idx] = g * (oo[r] * linv) + xb[idx];
        }
    }
}

// ---------------------------------------------------------------------------
extern "C" void kernel_launch(void* const* d_in, const int* in_sizes, int n_in,
                              void* d_out, int out_size, void* d_ws, size_t ws_size,
                              hipStream_t stream)
{
    const float* x     = (const float*)d_in[0];
    const float* Wq    = (const float*)d_in[1];
    const float* bq    = (const float*)d_in[2];
    const float* Wk    = (const float*)d_in[3];
    const float* bk    = (const float*)d_in[4];
    const float* Wv    = (const float*)d_in[5];
    const float* bv    = (const float*)d_in[6];
    const float* gamma = (const float*)d_in[7];

    float* ws = (float*)d_ws;
    float* q  = ws;                                // B*QK*N floats (512 KB)
    float* k  = q + (size_t)B_ * QK_ * N_;         // B*QK*N floats (512 KB)
    float* v  = k + (size_t)B_ * QK_ * N_;         // B*C*N floats  (4 MB)

    qkv_kernel<<<(B_ * N_) / 256, 256, 0, stream>>>(x, Wq, bq, Wk, bk, Wv, bv,
                                                    q, k, v);

    const int tiles = B_ * (N_ / 16);              // 1024 query tiles
    attn_kernel<<<tiles / 8, 256, 0, stream>>>(q, k, v, x, gamma,
                                               (float*)d_out);
}